// GCN_60722247631127
// MI455X (gfx1250) — compile-verified
//
#include <hip/hip_runtime.h>
#include <math.h>

// Problem constants (from reference)
#define NN   50000   // nodes
#define NE   625000  // edges
#define FIN  128
#define DH   128
#define NC   40
#define NCP  48      // NC padded to multiple of 16

typedef __attribute__((ext_vector_type(2))) float v2f;
typedef __attribute__((ext_vector_type(8))) float v8f;

static inline int cdiv(long a, long b) { return (int)((a + b - 1) / b); }

// ---------------- degree / norm ----------------

__global__ void k_deg_init(float* __restrict__ deg) {
    int i = blockIdx.x * blockDim.x + threadIdx.x;
    if (i < NN) deg[i] = 1.0f;   // self-loop contributes 1
}

__global__ void k_deg_accum(const int* __restrict__ dst, float* __restrict__ deg) {
    int e = blockIdx.x * blockDim.x + threadIdx.x;
    if (e < NE) atomicAdd(&deg[dst[e]], 1.0f);
}

__global__ void k_dinv(const float* __restrict__ deg, float* __restrict__ dinv) {
    int i = blockIdx.x * blockDim.x + threadIdx.x;
    if (i < NN) dinv[i] = rsqrtf(deg[i]);
}

__global__ void k_norm(const int* __restrict__ src, const int* __restrict__ dst,
                       const float* __restrict__ dinv, float* __restrict__ nrm) {
    int e = blockIdx.x * blockDim.x + threadIdx.x;
    if (e < NE) nrm[e] = dinv[src[e]] * dinv[dst[e]];
}

// zero-pad W2 [DH x NC] -> W2p [DH x NCP]
__global__ void k_pad_w2(const float* __restrict__ W2, float* __restrict__ W2p) {
    int t = blockIdx.x * blockDim.x + threadIdx.x;
    if (t >= DH * NCP) return;
    int r = t / NCP, c = t - r * NCP;
    W2p[t] = (c < NC) ? W2[r * NC + c] : 0.0f;
}

// ---------------- fp32 WMMA GEMM: C[M,NOUT] = A[M,K] @ B[K,NP] ----------------
// B (the weight matrix, K*NP floats) is staged once per block into LDS; every
// wave computes one 16x16 output tile with a fully-unrolled chain of
// V_WMMA_F32_16X16X4_F32. All loads are unconditional; only the final store
// is predicated (NP is padded, NOUT is the logical column count).

template <int K, int NP, int NOUT>
__global__ void k_gemm_wmma_f32(const float* __restrict__ A, const float* __restrict__ B,
                                float* __restrict__ C, int M) {
    __shared__ float ldsB[K * NP];

    // cooperative stage of B into LDS (float4, fully coalesced)
    for (int idx = threadIdx.x; idx < (K * NP) / 4; idx += blockDim.x)
        ((float4*)ldsB)[idx] = ((const float4*)B)[idx];
    __syncthreads();

    const int lane = threadIdx.x & 31;
    const int wave = (blockIdx.x * blockDim.x + threadIdx.x) >> 5;
    constexpr int tilesN = NP / 16;
    int tm = wave / tilesN;
    const int tn = wave - tm * tilesN;
    const bool valid = (tm * 16 < M);
    if (!valid) tm = M / 16 - 1;        // clamp tail waves (select, EXEC stays full)

    const int m  = tm * 16 + (lane & 15);   // A row for this lane
    const int n  = tn * 16 + (lane & 15);   // B col for this lane
    const int kh = (lane >> 4) << 1;        // lanes 16-31 hold K+2,K+3

    const float* aP   = A + m * K + kh;
    const int    bOff = kh * NP + n;

    v8f acc = {};
#pragma unroll
    for (int kk = 0; kk < K; kk += 4) {
        v2f a, b;
        a.x = aP[kk];
        a.y = aP[kk + 1];
        b.x = ldsB[bOff + kk * NP];
        b.y = ldsB[bOff + (kk + 1) * NP];
        acc = __builtin_amdgcn_wmma_f32_16x16x4_f32(
            /*neg_a=*/false, a, /*neg_b=*/false, b,
            /*c_mod=*/(short)0, acc, /*reuse_a=*/false, /*reuse_b=*/false);
    }

    if (valid && n < NOUT) {
        const int rbase = tm * 16 + ((lane >> 4) << 3);   // acc[v] -> row rbase+v
#pragma unroll
        for (int v = 0; v < 8; ++v)
            C[(rbase + v) * NOUT + n] = acc[v];
    }
}

// ---------------- aggregation, layer 1 (128 features) ----------------

// self-loop term initializes agg (no atomics needed)
__global__ void k_agg_init_h(const float* __restrict__ h, const float* __restrict__ dinv,
                             float* __restrict__ agg) {
    long t = (long)blockIdx.x * blockDim.x + threadIdx.x;   // NN*32 threads, float4 each
    if (t >= (long)NN * 32) return;
    int i = (int)(t >> 5);
    float s = dinv[i]; s *= s;
    float4 v = ((const float4*)h)[t];
    float4 o = make_float4(v.x * s, v.y * s, v.z * s, v.w * s);
    ((float4*)agg)[t] = o;
}

// one wave per edge, each lane owns 4 consecutive features
__global__ void k_agg_edges_h(const int* __restrict__ src, const int* __restrict__ dst,
                              const float* __restrict__ nrm,
                              const float* __restrict__ h, float* __restrict__ agg) {
    long t = (long)blockIdx.x * blockDim.x + threadIdx.x;
    long e = t >> 5;
    int lane = (int)(t & 31);
    if (e >= NE) return;
    int s = src[e], d = dst[e];
    float w = nrm[e];
    float4 v = ((const float4*)h)[(long)s * 32 + lane];
    float* o = agg + (long)d * 128 + lane * 4;
    atomicAdd(o + 0, v.x * w);
    atomicAdd(o + 1, v.y * w);
    atomicAdd(o + 2, v.z * w);
    atomicAdd(o + 3, v.w * w);
}

__global__ void k_bias_relu(float* __restrict__ agg, const float* __restrict__ b) {
    long t = (long)blockIdx.x * blockDim.x + threadIdx.x;
    if (t >= (long)NN * DH) return;
    float v = agg[t] + b[t & (DH - 1)];
    agg[t] = v > 0.0f ? v : 0.0f;
}

// ---------------- aggregation, layer 2 (40 features) ----------------

__global__ void k_agg_init_c(const float* __restrict__ h, const float* __restrict__ dinv,
                             float* __restrict__ agg) {
    long t = (long)blockIdx.x * blockDim.x + threadIdx.x;
    if (t >= (long)NN * NC) return;
    int i = (int)(t / NC);
    float s = dinv[i]; s *= s;
    agg[t] = h[t] * s;
}

__global__ void k_agg_edges_c(const int* __restrict__ src, const int* __restrict__ dst,
                              const float* __restrict__ nrm,
                              const float* __restrict__ h, float* __restrict__ agg) {
    long t = (long)blockIdx.x * blockDim.x + threadIdx.x;
    long e = t >> 5;
    int lane = (int)(t & 31);
    if (e >= NE) return;
    int s = src[e], d = dst[e];
    float w = nrm[e];
    atomicAdd(&agg[(long)d * NC + lane], h[(long)s * NC + lane] * w);
    if (lane < NC - 32) {
        int c = lane + 32;
        atomicAdd(&agg[(long)d * NC + c], h[(long)s * NC + c] * w);
    }
}

// ---------------- log_softmax over 40 classes ----------------

__global__ void k_logsoftmax(const float* __restrict__ agg, const float* __restrict__ b,
                             float* __restrict__ out) {
    int i = blockIdx.x * blockDim.x + threadIdx.x;
    if (i >= NN) return;
    const float* r = agg + (long)i * NC;
    float mx = -INFINITY;
#pragma unroll
    for (int c = 0; c < NC; ++c) mx = fmaxf(mx, r[c] + b[c]);
    float sum = 0.0f;
#pragma unroll
    for (int c = 0; c < NC; ++c) sum += expf(r[c] + b[c] - mx);
    float lse = mx + logf(sum);
#pragma unroll
    for (int c = 0; c < NC; ++c) out[(long)i * NC + c] = r[c] + b[c] - lse;
}

// ---------------- launch ----------------

extern "C" void kernel_launch(void* const* d_in, const int* in_sizes, int n_in,
                              void* d_out, int out_size, void* d_ws, size_t ws_size,
                              hipStream_t stream) {
    const float* x  = (const float*)d_in[0];            // [NN, FIN]
    const int*   ei = (const int*)d_in[1];              // [2, NE]
    const float* W1 = (const float*)d_in[2];            // [FIN, DH]
    const float* b1 = (const float*)d_in[3];            // [DH]
    const float* W2 = (const float*)d_in[4];            // [DH, NC]
    const float* b2 = (const float*)d_in[5];            // [NC]
    float* out = (float*)d_out;                          // [NN, NC]

    const int* src = ei;
    const int* dst = ei + NE;

    // workspace layout (floats), all 16B-aligned
    float* ws   = (float*)d_ws;
    float* deg  = ws;                          // NN
    float* dinv = deg  + NN;                   // NN
    float* nrm  = dinv + NN;                   // NE
    float* h0   = nrm  + NE;                   // NN*DH
    float* agg1 = h0   + (long)NN * DH;        // NN*DH  (becomes h1 after bias+relu)
    float* h2   = agg1 + (long)NN * DH;        // NN*NC
    float* agg2 = h2   + (long)NN * NC;        // NN*NC
    float* W2p  = agg2 + (long)NN * NC;        // DH*NCP (zero-padded W2)

    const int B = 256;

    // normalization
    k_deg_init <<<cdiv(NN, B), B, 0, stream>>>(deg);
    k_deg_accum<<<cdiv(NE, B), B, 0, stream>>>(dst, deg);
    k_dinv     <<<cdiv(NN, B), B, 0, stream>>>(deg, dinv);
    k_norm     <<<cdiv(NE, B), B, 0, stream>>>(src, dst, dinv, nrm);
    k_pad_w2   <<<cdiv(DH * NCP, B), B, 0, stream>>>(W2, W2p);

    // layer 1: h0 = x @ W1   (3125 x 8 tiles; exact grid, no tail)
    {
        long waves = (long)(NN / 16) * (DH / 16);
        k_gemm_wmma_f32<FIN, DH, DH>
            <<<cdiv(waves * 32, B), B, 0, stream>>>(x, W1, h0, NN);
    }
    k_agg_init_h <<<cdiv((long)NN * 32, B), B, 0, stream>>>(h0, dinv, agg1);
    k_agg_edges_h<<<cdiv((long)NE * 32, B), B, 0, stream>>>(src, dst, nrm, h0, agg1);
    k_bias_relu  <<<cdiv((long)NN * DH, B), B, 0, stream>>>(agg1, b1);

    // layer 2: h2 = h1 @ W2p  (3125 x 3 tiles; tail waves clamp + masked store)
    {
        long waves = (long)(NN / 16) * (NCP / 16);
        k_gemm_wmma_f32<DH, NCP, NC>
            <<<cdiv(waves * 32, B), B, 0, stream>>>(agg1, W2p, h2, NN);
    }
    k_agg_init_c <<<cdiv((long)NN * NC, B), B, 0, stream>>>(h2, dinv, agg2);
    k_agg_edges_c<<<cdiv((long)NE * 32, B), B, 0, stream>>>(src, dst, nrm, h2, agg2);

    // output
    k_logsoftmax<<<cdiv(NN, B), B, 0, stream>>>(agg2, b2, out);
}